// AdvancedFuser_77438260347299
// MI455X (gfx1250) — compile-verified
//
#include <hip/hip_runtime.h>
#include <math.h>

// MI455X / gfx1250, wave32. Memory-bound fusion kernel:
//   weighted = softmax(rw) . retrieved ; gate = sigmoid([q;weighted] @ gate_w + b)
//   fused = gate*q + (1-gate)*weighted
// Gate GEMV is done with V_WMMA_F32_16X16X4_F32 out of LDS (f32 end-to-end).

typedef __attribute__((ext_vector_type(2))) float v2f;
typedef __attribute__((ext_vector_type(4))) float v4f;
typedef __attribute__((ext_vector_type(8))) float v8f;

#define EMBED    512
#define KRETR    5
#define ROWS     16    // rows per tile == WMMA N
#define THREADS  128   // 4 waves per workgroup
#define NWAVES   4
#define XSTRIDE  1028  // 1024 payload + 4 pad; stride % 64 == 4 -> conflict-free phase-B b64 reads

__global__ __launch_bounds__(THREADS)
void AdvancedFuser_kernel(const float* __restrict__ q,
                          const float* __restrict__ retr,
                          const float* __restrict__ rw,
                          const float* __restrict__ gw,
                          const float* __restrict__ gb,
                          float* __restrict__ out)
{
    // Per row r: sX[r*XSTRIDE + 0..511] = q row, [512..1023] = weighted row.
    __shared__ __align__(16) float sX[ROWS * XSTRIDE];
    __shared__ __align__(16) float sGW[2 * EMBED];
    __shared__ float sPart[NWAVES * ROWS];
    __shared__ float sGate[ROWS];

    const int t    = threadIdx.x;
    const int wave = t >> 5;
    const int lane = t & 31;
    const int n    = lane & 15;   // WMMA B/D column = batch row within tile
    const int hi   = lane >> 4;   // lane half selects K half within a K=4 chunk
    const long row0 = (long)blockIdx.x * ROWS;

    // softmax over the 5 slot weights (20B, cached; every thread redundantly)
    float e0 = rw[0], e1 = rw[1], e2 = rw[2], e3 = rw[3], e4 = rw[4];
    const float mx = fmaxf(fmaxf(fmaxf(e0, e1), fmaxf(e2, e3)), e4);
    e0 = __expf(e0 - mx); e1 = __expf(e1 - mx); e2 = __expf(e2 - mx);
    e3 = __expf(e3 - mx); e4 = __expf(e4 - mx);
    const float inv = 1.0f / (e0 + e1 + e2 + e3 + e4);
    const float w0 = e0 * inv, w1 = e1 * inv, w2 = e2 * inv, w3 = e3 * inv, w4 = e4 * inv;

    // Stage gate_w (4KB) into LDS once; reused by all WMMA steps.
    {
        const v4f* g4 = (const v4f*)gw;
        ((v4f*)sGW)[t]           = g4[t];
        ((v4f*)sGW)[THREADS + t] = g4[THREADS + t];
    }

    // ---- Phase A: stream q + retrieved once (NT), build weighted, stage tile in LDS.
    const int d = 4 * t;  // 128 threads * float4 == one 512-float row per iteration
    for (int r = 0; r < ROWS; ++r) {
        const long row = row0 + r;
        const v4f q4 = __builtin_nontemporal_load((const v4f*)(q + row * EMBED + d));
        const float* rb = retr + (row * KRETR) * EMBED + d;
        const v4f a0 = __builtin_nontemporal_load((const v4f*)(rb + 0 * EMBED));
        const v4f a1 = __builtin_nontemporal_load((const v4f*)(rb + 1 * EMBED));
        const v4f a2 = __builtin_nontemporal_load((const v4f*)(rb + 2 * EMBED));
        const v4f a3 = __builtin_nontemporal_load((const v4f*)(rb + 3 * EMBED));
        const v4f a4 = __builtin_nontemporal_load((const v4f*)(rb + 4 * EMBED));
        const v4f ws = w0 * a0 + w1 * a1 + w2 * a2 + w3 * a3 + w4 * a4;
        *(v4f*)(sX + r * XSTRIDE + d)         = q4;
        *(v4f*)(sX + r * XSTRIDE + EMBED + d) = ws;
    }
    __syncthreads();

    // ---- Phase B: gate logits via V_WMMA_F32_16X16X4_F32.
    // A(16x4): A[m][k] = gw[kc+k] (same for all m). ISA A layout: lanes 0-15 hold
    //   K = vgpr (0,1); lanes 16-31 hold K = 2+vgpr.  -> a = gw[kc + 2*hi + {0,1}]
    // B(4x16): B[k][n] = X[n][kc+k]. ISA B layout (half-split K, per SWMMAC B docs):
    //   lanes 0-15 hold K = {0,1}, lanes 16-31 hold K = {2,3} across the 2 VGPRs.
    //   -> b = X[n][kc + 2*hi + {0,1}]  (one ds_load_b64, all 64 banks hit once)
    // Then D[m][n] = sum_k gw[kc+k]*X[n][kc+k], identical for every m.
    v8f acc = {};
    const int kbase = wave * (2 * EMBED / NWAVES);  // 256 K per wave
    #pragma unroll 4
    for (int i = 0; i < 64; ++i) {
        const int kc = kbase + i * 4;
        const v2f a = *(const v2f*)(sGW + kc + 2 * hi);
        const v2f b = *(const v2f*)(sX + n * XSTRIDE + kc + 2 * hi);
        acc = __builtin_amdgcn_wmma_f32_16x16x4_f32(
            /*neg_a=*/false, a, /*neg_b=*/false, b,
            /*c_mod=*/(short)0, acc, /*reuse_a=*/false, /*reuse_b=*/false);
    }
    if (lane < 16) sPart[wave * ROWS + lane] = acc[0];  // all D rows equal; take M=0
    __syncthreads();

    if (t < ROWS) {
        const float s = sPart[t] + sPart[ROWS + t] + sPart[2 * ROWS + t] +
                        sPart[3 * ROWS + t] + gb[0];
        sGate[t] = 1.0f / (1.0f + __expf(-s));
    }
    __syncthreads();

    // ---- Phase C: fused = g*q + (1-g)*weighted, from LDS, NT stores.
    for (int r = 0; r < ROWS; ++r) {
        const float g = sGate[r];
        const v4f q4 = *(const v4f*)(sX + r * XSTRIDE + d);
        const v4f wv = *(const v4f*)(sX + r * XSTRIDE + EMBED + d);
        const v4f o  = g * q4 + (1.0f - g) * wv;
        __builtin_nontemporal_store(o, (v4f*)(out + (row0 + r) * EMBED + d));
    }
}

extern "C" void kernel_launch(void* const* d_in, const int* in_sizes, int n_in,
                              void* d_out, int out_size, void* d_ws, size_t ws_size,
                              hipStream_t stream) {
    const float* q    = (const float*)d_in[0];  // (B, 512)
    const float* retr = (const float*)d_in[1];  // (B, 5, 512)
    const float* rw   = (const float*)d_in[2];  // (5,)
    const float* gw   = (const float*)d_in[3];  // (1, 1024)
    const float* gb   = (const float*)d_in[4];  // (1,)
    float* out = (float*)d_out;                 // (B, 512)

    const int nrows = in_sizes[0] / EMBED;      // 65536 (multiple of 16)
    const int tiles = nrows / ROWS;
    AdvancedFuser_kernel<<<tiles, THREADS, 0, stream>>>(q, retr, rw, gw, gb, out);
}